// TransformerBlockWithCache_15375982919772
// MI455X (gfx1250) — compile-verified
//
#include <hip/hip_runtime.h>
#include <hip/hip_bf16.h>

#define E    2048
#define NH   16
#define HD   128
#define KVC  8192
#define KV1  8193
#define SPAD 8208   // KV1 padded to multiple of 16
#define FF   128
#define EPSLN 1e-6f

typedef __attribute__((ext_vector_type(2))) float v2f;
typedef __attribute__((ext_vector_type(8))) float v8f;
typedef int v4i __attribute__((vector_size(4 * sizeof(int))));

#if defined(__has_builtin)
#  if __has_builtin(__builtin_amdgcn_global_load_async_to_lds_b128) && \
      __has_builtin(__builtin_amdgcn_s_wait_asynccnt)
#    define USE_ASYNC_LDS 1
#  endif
#endif

__device__ __forceinline__ v8f wmma_f32x4(v2f a, v2f b, v8f c) {
  // D = A(16x4 f32) * B(4x16 f32) + C(16x16 f32)
  return __builtin_amdgcn_wmma_f32_16x16x4_f32(false, a, false, b, (short)0, c,
                                               false, false);
}

// Stage K floats (K % 4 == 0) from global into LDS, block-cooperative.
__device__ __forceinline__ void stage_to_lds(float* xs, const float* __restrict__ xin,
                                             int K, int tid, int nthr) {
#ifdef USE_ASYNC_LDS
  int nv4 = K >> 2;
  for (int i = tid; i < nv4; i += nthr) {
    __builtin_amdgcn_global_load_async_to_lds_b128(
        (__attribute__((address_space(1))) v4i*)(xin + i * 4),
        (__attribute__((address_space(3))) v4i*)(xs + i * 4), 0, 0);
  }
  __builtin_amdgcn_s_wait_asynccnt(0);
#else
  int nv4 = K >> 2;
  const float4* src = (const float4*)xin;
  float4* dst = (float4*)xs;
  for (int i = tid; i < nv4; i += nthr) dst[i] = src[i];
#endif
  __syncthreads();
}

// out[n] = sum_k xin[k]*W[k*N+n] + bias[n]  (optionally relu)
// One wave per 16-column tile; A = x chunk broadcast across M rows.
__global__ void gemv_wmma(const float* __restrict__ W, const float* __restrict__ xin,
                          const float* __restrict__ bias, float* __restrict__ out,
                          int K, int N, int relu) {
  __shared__ float xs[E];
  int tid = threadIdx.x;
  stage_to_lds(xs, xin, K, tid, blockDim.x);

  int lane = tid & 31, wave = tid >> 5;
  int tile = blockIdx.x * (blockDim.x >> 5) + wave;
  if (tile * 16 >= N) return;                 // wave-uniform
  int n0 = tile * 16;
  int h2 = (lane >> 4) * 2;                   // K sub-offset per half-wave
  int nl = lane & 15;

  v8f c = {0.f,0.f,0.f,0.f,0.f,0.f,0.f,0.f};
  for (int kc = 0; kc < K; kc += 4) {
    v2f a, b;
    a.x = xs[kc + h2];
    a.y = xs[kc + h2 + 1];
    const float* wp = W + (size_t)(kc + h2) * N + (n0 + nl);
    b.x = wp[0];
    b.y = wp[N];
    if (kc + 16 < K) __builtin_prefetch(wp + (size_t)16 * N, 0, 0);
    c = wmma_f32x4(a, b, c);
  }
  if (lane < 16) {                            // D row 0 = lanes 0..15, VGPR0
    float r = c[0] + bias[n0 + lane];
    if (relu) r = fmaxf(r, 0.0f);
    out[n0 + lane] = r;
  }
}

// One block (256 thr = 8 waves) per head. Scores/probabilities live in LDS
// (SPAD*4 = 32.8 KB of the WGP's 320 KB) so vmem is only the one-pass K/V stream.
// Phase A: scores via WMMA (K-cache rows = M dim). Softmax reduce in LDS.
// Phase B: out = p @ V via WMMA (wave w -> d-tile w*16).
__global__ void attn_wmma(const float* __restrict__ kmat,   // [KV1, NH, HD]
                          const float* __restrict__ vmat,   // [KV1, NH, HD]
                          const float* __restrict__ q,      // [NH*HD]
                          float* __restrict__ values) {     // [NH*HD]
  int h = blockIdx.x;
  int tid = threadIdx.x;
  int lane = tid & 31, wave = tid >> 5;
  __shared__ float ph[SPAD];
  __shared__ float qs[HD];
  __shared__ float red[256];
  __shared__ float stat[2];
  if (tid < HD) qs[tid] = q[h * HD + tid];
  __syncthreads();

  const float SC = 0.08838834764831845f;      // 1/sqrt(128)
  int h2 = (lane >> 4) * 2;
  int m  = lane & 15;

  // ---- Phase A: scores[j] = (K[j,h,:] . q_h) * SC ----
  const int NT = (KV1 + 15) / 16;             // 513 tiles of 16 rows
  for (int t = wave; t < NT; t += 8) {
    int j0 = t * 16;
    int j = j0 + m; if (j > KVC) j = KVC;     // clamp tail loads
    const float* kp = kmat + (size_t)j * E + h * HD;
    v8f c = {0.f,0.f,0.f,0.f,0.f,0.f,0.f,0.f};
    for (int kc = 0; kc < HD; kc += 4) {
      v2f a, b;
      a.x = kp[kc + h2];
      a.y = kp[kc + h2 + 1];
      b.x = qs[kc + h2];                      // broadcast q across N
      b.y = qs[kc + h2 + 1];
      c = wmma_f32x4(a, b, c);
    }
    if (m == 0) {                             // lanes 0 & 16 hold column N=0
      int mb = (lane >> 4) * 8;
      #pragma unroll
      for (int r = 0; r < 8; ++r) {
        int jj = j0 + mb + r;
        if (jj < KV1) ph[jj] = c[r] * SC;
      }
    }
  }
  __syncthreads();

  // ---- softmax: max ----
  float lm = -__builtin_inff();
  for (int j = tid; j < KV1; j += 256) lm = fmaxf(lm, ph[j]);
  red[tid] = lm; __syncthreads();
  for (int s = 128; s > 0; s >>= 1) {
    if (tid < s) red[tid] = fmaxf(red[tid], red[tid + s]);
    __syncthreads();
  }
  if (tid == 0) stat[0] = red[0];
  __syncthreads();
  float smax = stat[0];
  // ---- softmax: exp + sum, zero the pad ----
  float ls = 0.f;
  for (int j = tid; j < KV1; j += 256) {
    float e = __expf(ph[j] - smax);
    ph[j] = e; ls += e;
  }
  if (tid < SPAD - KV1) ph[KV1 + tid] = 0.f;
  red[tid] = ls; __syncthreads();
  for (int s = 128; s > 0; s >>= 1) {
    if (tid < s) red[tid] += red[tid + s];
    __syncthreads();
  }
  if (tid == 0) stat[1] = 1.0f / red[0];
  __syncthreads();
  float inv = stat[1];

  // ---- Phase B: values[h, d0+n] = sum_j p[j] * V[j,h,d0+n] ----
  int d0 = wave * 16;                         // 8 waves x 16 = 128 = HD
  int nl = lane & 15;
  v8f c = {0.f,0.f,0.f,0.f,0.f,0.f,0.f,0.f};
  for (int j0 = 0; j0 < SPAD; j0 += 4) {
    v2f a, b;
    a.x = ph[j0 + h2];                        // broadcast p across M (LDS)
    a.y = ph[j0 + h2 + 1];
    const float* vp = vmat + (size_t)(j0 + h2) * E + h * HD + d0 + nl;
    b.x = vp[0];
    b.y = vp[E];
    c = wmma_f32x4(a, b, c);
  }
  if (lane < 16) values[h * HD + d0 + lane] = c[0] * inv;
}

// out = LayerNorm(a + b) * sc + bi   over E elements, one block of 256
__global__ void add_ln(const float* __restrict__ a, const float* __restrict__ b,
                       const float* __restrict__ sc, const float* __restrict__ bi,
                       float* __restrict__ out) {
  __shared__ float buf[E];
  __shared__ float rs[256], rq[256];
  int tid = threadIdx.x;
  float s = 0.f, qq = 0.f;
  const float4* a4 = (const float4*)a;
  const float4* b4 = (const float4*)b;
  float4* buf4 = (float4*)buf;
  for (int i = tid; i < E / 4; i += 256) {
    float4 av = a4[i], bv = b4[i];
    float4 t;
    t.x = av.x + bv.x; t.y = av.y + bv.y; t.z = av.z + bv.z; t.w = av.w + bv.w;
    buf4[i] = t;
    s += t.x + t.y + t.z + t.w;
    qq += t.x * t.x + t.y * t.y + t.z * t.z + t.w * t.w;
  }
  rs[tid] = s; rq[tid] = qq; __syncthreads();
  for (int st = 128; st > 0; st >>= 1) {
    if (tid < st) { rs[tid] += rs[tid + st]; rq[tid] += rq[tid + st]; }
    __syncthreads();
  }
  float mean = rs[0] * (1.0f / E);
  float var  = rq[0] * (1.0f / E) - mean * mean;
  float invs = rsqrtf(var + EPSLN);
  const float4* sc4 = (const float4*)sc;
  const float4* bi4 = (const float4*)bi;
  float4* out4 = (float4*)out;
  for (int i = tid; i < E / 4; i += 256) {
    float4 t = buf4[i], sv = sc4[i], bv = bi4[i];
    float4 o;
    o.x = (t.x - mean) * invs * sv.x + bv.x;
    o.y = (t.y - mean) * invs * sv.y + bv.y;
    o.z = (t.z - mean) * invs * sv.z + bv.z;
    o.w = (t.w - mean) * invs * sv.w + bv.w;
    out4[i] = o;
  }
}

extern "C" void kernel_launch(void* const* d_in, const int* in_sizes, int n_in,
                              void* d_out, int out_size, void* d_ws, size_t ws_size,
                              hipStream_t stream) {
  const float* x       = (const float*)d_in[0];
  const float* v_cache = (const float*)d_in[1];
  const float* k_cache = (const float*)d_in[2];
  const float* Wv = (const float*)d_in[3];  const float* bv = (const float*)d_in[4];
  const float* Wq = (const float*)d_in[5];  const float* bq = (const float*)d_in[6];
  const float* Wk = (const float*)d_in[7];  const float* bk = (const float*)d_in[8];
  const float* Wo = (const float*)d_in[9];  const float* bo = (const float*)d_in[10];
  const float* W1 = (const float*)d_in[11]; const float* b1 = (const float*)d_in[12];
  const float* W2 = (const float*)d_in[13]; const float* b2 = (const float*)d_in[14];
  const float* ln1s = (const float*)d_in[15]; const float* ln1b = (const float*)d_in[16];
  const float* ln2s = (const float*)d_in[17]; const float* ln2b = (const float*)d_in[18];

  float* out_x = (float*)d_out;                       // [E]
  float* out_v = out_x + E;                           // [KV1, NH, HD]
  float* out_k = out_v + (size_t)KV1 * E;             // [KV1, NH, HD]

  float* ws   = (float*)d_ws;
  float* qv   = ws;                // E
  float* vals = ws + E;            // E
  float* ao   = ws + 2 * E;        // E
  float* x1   = ws + 3 * E;        // E
  float* hb   = ws + 4 * E;        // FF (128)
  float* ffo  = ws + 4 * E + 256;  // E

  // Cache concat: bulk D2D copies (pure bandwidth, ~134 MB)
  (void)hipMemcpyAsync(out_v, v_cache, (size_t)KVC * E * sizeof(float),
                       hipMemcpyDeviceToDevice, stream);
  (void)hipMemcpyAsync(out_k, k_cache, (size_t)KVC * E * sizeof(float),
                       hipMemcpyDeviceToDevice, stream);

  // q/k/v projections (k_i, v_i written directly into appended cache row)
  gemv_wmma<<<16, 256, 0, stream>>>(Wq, x, bq, qv, E, E, 0);
  gemv_wmma<<<16, 256, 0, stream>>>(Wk, x, bk, out_k + (size_t)KVC * E, E, E, 0);
  gemv_wmma<<<16, 256, 0, stream>>>(Wv, x, bv, out_v + (size_t)KVC * E, E, E, 0);

  // attention (one block per head)
  attn_wmma<<<NH, 256, 0, stream>>>(out_k, out_v, qv, vals);

  // output projection + residual + LN1
  gemv_wmma<<<16, 256, 0, stream>>>(Wo, vals, bo, ao, E, E, 0);
  add_ln<<<1, 256, 0, stream>>>(x, ao, ln1s, ln1b, x1);

  // FFN + residual + LN2
  gemv_wmma<<<1, 256, 0, stream>>>(W1, x1, b1, hb, E, FF, 1);
  gemv_wmma<<<16, 256, 0, stream>>>(W2, hb, b2, ffo, FF, E, 0);
  add_ln<<<1, 256, 0, stream>>>(x1, ffo, ln2s, ln2b, out_x);
}